// GAT_69647189671984
// MI455X (gfx1250) — compile-verified
//
#include <hip/hip_runtime.h>
#include <math.h>

// Problem constants (must match reference)
#define NN   50000
#define EE   400000
#define HH   4
#define DD   64
#define FF   256        // H*D, also IN for layer0
#define NEG_SLOPE 0.2f

typedef __attribute__((ext_vector_type(2))) float v2f;
typedef __attribute__((ext_vector_type(8))) float v8f;
typedef __attribute__((ext_vector_type(4))) int   v4i;

// CDNA5 async global->LDS staging (ASYNCcnt-tracked, bypasses VGPRs).
#if __has_builtin(__builtin_amdgcn_global_load_async_to_lds_b128) && \
    __has_builtin(__builtin_amdgcn_s_wait_asynccnt)
#define USE_ASYNC_LDS 1
#else
#define USE_ASYNC_LDS 0
#endif

// ---------------------------------------------------------------------------
// Utility: fill buffer with a constant
// ---------------------------------------------------------------------------
__global__ void fill_kernel(float* __restrict__ p, float v, int n) {
    int t = blockIdx.x * blockDim.x + threadIdx.x;
    if (t < n) p[t] = v;
}

// ---------------------------------------------------------------------------
// WMMA fp32 GEMM:  C[m,n] = sum_k A[m,k] * W[n,k]
// A: [M,K] row-major, W: [Nw,K] row-major (torch Linear layout), C: [M,Nw]
// Tile 64x64 per 256-thread block, K chunked by 64 through LDS.
// 8 waves/block; each wave owns a 32(M) x 16(N) subtile via two accumulators.
// Staging uses GLOBAL_LOAD_ASYNC_TO_LDS_B128 when available.
// ---------------------------------------------------------------------------
#define TM 64
#define TN 64
#define TK 64
#define LDP 68   // padded LDS row stride (floats): 16B-aligned rows, bank-shifted

__global__ __launch_bounds__(256) void gemm_nt_wmma(
    const float* __restrict__ A, const float* __restrict__ W,
    float* __restrict__ C, int M, int Nw, int K)
{
    __shared__ float sA[TM][LDP];
    __shared__ float sB[TN][LDP];

    const int bm   = blockIdx.x * TM;
    const int bn   = blockIdx.y * TN;
    const int tid  = threadIdx.x;
    const int wave = tid >> 5;
    const int lane = tid & 31;

    const int wn  = (wave & 3) * 16;   // N offset of this wave's subtile
    const int wm0 = (wave >> 2) * 32;  // M offset (covers 32 rows via 2 accs)

    v8f acc0 = {};
    v8f acc1 = {};

    const int mrow = lane & 15;
    const int kh   = (lane >> 4) * 2;  // lanes 0-15 -> K 0/1, lanes 16-31 -> K 2/3

    for (int k0 = 0; k0 < K; k0 += TK) {
        // ---- stage A and W tiles into LDS ----
#if USE_ASYNC_LDS
        for (int i = tid; i < TM * (TK / 4); i += 256) {
            int r  = i / (TK / 4);
            int c4 = (i % (TK / 4)) * 4;
            // clamp OOB rows to a valid row: garbage only feeds C rows >= M,
            // which are never stored.
            int arow = bm + r; arow = arow < M  ? arow : M  - 1;
            int brow = bn + r; brow = brow < Nw ? brow : Nw - 1;
            __builtin_amdgcn_global_load_async_to_lds_b128(
                (v4i*)&A[(size_t)arow * K + k0 + c4],
                (v4i*)&sA[r][c4], 0, 0);
            __builtin_amdgcn_global_load_async_to_lds_b128(
                (v4i*)&W[(size_t)brow * K + k0 + c4],
                (v4i*)&sB[r][c4], 0, 0);
        }
        __builtin_amdgcn_s_wait_asynccnt(0);
        __syncthreads();
#else
        for (int i = tid; i < TM * (TK / 4); i += 256) {
            int r  = i / (TK / 4);
            int c4 = (i % (TK / 4)) * 4;
            float4 av = make_float4(0.f, 0.f, 0.f, 0.f);
            if (bm + r < M)
                av = *(const float4*)&A[(size_t)(bm + r) * K + k0 + c4];
            *(float4*)&sA[r][c4] = av;
            float4 bv = make_float4(0.f, 0.f, 0.f, 0.f);
            if (bn + r < Nw)
                bv = *(const float4*)&W[(size_t)(bn + r) * K + k0 + c4];
            *(float4*)&sB[r][c4] = bv;
        }
        __syncthreads();
#endif

        // ---- 16 wmma steps over this K chunk ----
        for (int kk = 0; kk < TK; kk += 4) {
            float2 ta0 = *(const float2*)&sA[wm0 + mrow][kk + kh];
            float2 ta1 = *(const float2*)&sA[wm0 + 16 + mrow][kk + kh];
            float2 tb  = *(const float2*)&sB[wn + mrow][kk + kh];
            v2f a0; a0.x = ta0.x; a0.y = ta0.y;
            v2f a1; a1.x = ta1.x; a1.y = ta1.y;
            v2f b;  b.x  = tb.x;  b.y  = tb.y;
            acc0 = __builtin_amdgcn_wmma_f32_16x16x4_f32(
                false, a0, false, b, (short)0, acc0, false, false);
            acc1 = __builtin_amdgcn_wmma_f32_16x16x4_f32(
                false, a1, false, b, (short)0, acc1, false, false);
        }
        __syncthreads();
    }

    // ---- write back: vgpr r -> M=r (lanes 0-15) or M=r+8 (lanes 16-31) ----
    const int cn     = bn + wn + (lane & 15);
    const int cm_off = (lane >> 4) * 8;
    if (cn < Nw) {
        for (int r = 0; r < 8; ++r) {
            int m0 = bm + wm0 + cm_off + r;
            int m1 = m0 + 16;
            if (m0 < M) C[(size_t)m0 * Nw + cn] = acc0[r];
            if (m1 < M) C[(size_t)m1 * Nw + cn] = acc1[r];
        }
    }
}

// ---------------------------------------------------------------------------
// Attention coefficients: el[n,h] = dot(feat[n,h,:], attn_l[h,:]); same for er.
// One wave32 per (n,h); lane handles 2 consecutive dims (float2).
// ---------------------------------------------------------------------------
__global__ __launch_bounds__(256) void elr_kernel(
    const float* __restrict__ feat, const float* __restrict__ al,
    const float* __restrict__ ar, float* __restrict__ el,
    float* __restrict__ er, int total /* = N*H */)
{
    int g    = blockIdx.x * 8 + (threadIdx.x >> 5);
    int lane = threadIdx.x & 31;
    if (g >= total) return;
    int n = g >> 2;
    int h = g & 3;
    float2 f = ((const float2*)(feat + (size_t)n * FF + h * DD))[lane];
    float2 a = ((const float2*)(al + h * DD))[lane];
    float2 b = ((const float2*)(ar + h * DD))[lane];
    float sl = f.x * a.x + f.y * a.y;
    float sr = f.x * b.x + f.y * b.y;
    for (int off = 16; off; off >>= 1) {
        sl += __shfl_down(sl, off, 32);
        sr += __shfl_down(sr, off, 32);
    }
    if (lane == 0) { el[g] = sl; er[g] = sr; }
}

// ---------------------------------------------------------------------------
// Float atomic max via int-bit ordering trick (valid for all finite floats,
// emax initialized to -inf).
// ---------------------------------------------------------------------------
__device__ __forceinline__ void atomicMaxF32(float* addr, float v) {
    if (v >= 0.0f)
        atomicMax((int*)addr, __float_as_int(v));
    else
        atomicMin((unsigned int*)addr, __float_as_uint(v));
}

__device__ __forceinline__ float edge_score(const float* el, const float* er,
                                            int s, int d, int h) {
    float v = el[s * HH + h] + er[d * HH + h];
    return v > 0.0f ? v : NEG_SLOPE * v;
}

// pass 1: per-dst running max of leaky-relu'd scores
__global__ void edge_max_kernel(const float* __restrict__ el,
                                const float* __restrict__ er,
                                const int* __restrict__ src,
                                const int* __restrict__ dst,
                                float* __restrict__ emax, int totalEH)
{
    int t = blockIdx.x * blockDim.x + threadIdx.x;
    if (t >= totalEH) return;
    int e = t >> 2, h = t & 3;
    int s = src[e], d = dst[e];
    atomicMaxF32(&emax[d * HH + h], edge_score(el, er, s, d, h));
}

// pass 2: ee = exp(e - emax[dst]); denom[dst] += ee
__global__ void edge_exp_kernel(const float* __restrict__ el,
                                const float* __restrict__ er,
                                const int* __restrict__ src,
                                const int* __restrict__ dst,
                                const float* __restrict__ emax,
                                float* __restrict__ ee,
                                float* __restrict__ denom, int totalEH)
{
    int t = blockIdx.x * blockDim.x + threadIdx.x;
    if (t >= totalEH) return;
    int e = t >> 2, h = t & 3;
    int s = src[e], d = dst[e];
    float v = __expf(edge_score(el, er, s, d, h) - emax[d * HH + h]);
    ee[t] = v;
    atomicAdd(&denom[d * HH + h], v);
}

// pass 3: rst[dst, h, :] += feat[src, h, :] * (ee / denom[dst,h])
// one block per edge; 256 threads = full H*D feature row (coalesced).
__global__ __launch_bounds__(256) void aggregate_kernel(
    const float* __restrict__ feat, const float* __restrict__ ee,
    const float* __restrict__ denom, const int* __restrict__ src,
    const int* __restrict__ dst, float* __restrict__ rst)
{
    int e = blockIdx.x;
    int c = threadIdx.x;          // h*64 + d
    int h = c >> 6;
    int s = src[e], d = dst[e];
    float a = ee[e * HH + h] / denom[d * HH + h];
    atomicAdd(&rst[(size_t)d * FF + c], feat[(size_t)s * FF + c] * a);
}

// finalize layer0: out = elu(rst + bias)
__global__ void finalize_elu_kernel(float* __restrict__ rst,
                                    const float* __restrict__ bias, int total)
{
    int t = blockIdx.x * blockDim.x + threadIdx.x;
    if (t >= total) return;
    float v = rst[t] + bias[t & (FF - 1)];
    rst[t] = v > 0.0f ? v : expm1f(v);
}

// finalize layer1: out = rst + residual + bias
__global__ void finalize_res_kernel(float* __restrict__ rst,
                                    const float* __restrict__ resid,
                                    const float* __restrict__ bias, int total)
{
    int t = blockIdx.x * blockDim.x + threadIdx.x;
    if (t >= total) return;
    rst[t] = rst[t] + resid[t] + bias[t & (FF - 1)];
}

// per-(edge,head) dot product over D of h[src]·h[dst]; wave32 reduction
__global__ __launch_bounds__(256) void score_kernel(
    const float* __restrict__ hf, const int* __restrict__ src,
    const int* __restrict__ dst, float* __restrict__ out, int totalEH)
{
    int g    = blockIdx.x * 8 + (threadIdx.x >> 5);
    int lane = threadIdx.x & 31;
    if (g >= totalEH) return;
    int e = g >> 2, h = g & 3;
    int s = src[e], d = dst[e];
    float2 a = ((const float2*)(hf + (size_t)s * FF + h * DD))[lane];
    float2 b = ((const float2*)(hf + (size_t)d * FF + h * DD))[lane];
    float v = a.x * b.x + a.y * b.y;
    for (int off = 16; off; off >>= 1) v += __shfl_down(v, off, 32);
    if (lane == 0) out[g] = v;
}

// ---------------------------------------------------------------------------
// Host-side orchestration
// ---------------------------------------------------------------------------
static void run_gat_layer(hipStream_t stream,
                          const float* hin, const float* W,
                          const float* al, const float* ar, const float* bias,
                          const int* src, const int* dst,
                          float* feat, float* el, float* er,
                          float* emax, float* denom, float* ee,
                          float* rst, const float* resid /* null -> elu */)
{
    const int NF  = NN * FF;
    const int NH  = NN * HH;
    const int EH  = EE * HH;

    // GEMM: feat = hin @ W^T
    dim3 ggrid((NN + TM - 1) / TM, FF / TN);
    hipLaunchKernelGGL(gemm_nt_wmma, ggrid, dim3(256), 0, stream,
                       hin, W, feat, NN, FF, FF);

    // attention coefficients
    hipLaunchKernelGGL(elr_kernel, dim3((NH + 7) / 8), dim3(256), 0, stream,
                       feat, al, ar, el, er, NH);

    // init softmax state + accumulator
    hipLaunchKernelGGL(fill_kernel, dim3((NH + 255) / 256), dim3(256), 0, stream,
                       emax, -INFINITY, NH);
    hipLaunchKernelGGL(fill_kernel, dim3((NH + 255) / 256), dim3(256), 0, stream,
                       denom, 0.0f, NH);
    hipLaunchKernelGGL(fill_kernel, dim3((NF + 255) / 256), dim3(256), 0, stream,
                       rst, 0.0f, NF);

    // edge softmax (3 passes) + aggregation
    hipLaunchKernelGGL(edge_max_kernel, dim3((EH + 255) / 256), dim3(256), 0,
                       stream, el, er, src, dst, emax, EH);
    hipLaunchKernelGGL(edge_exp_kernel, dim3((EH + 255) / 256), dim3(256), 0,
                       stream, el, er, src, dst, emax, ee, denom, EH);
    hipLaunchKernelGGL(aggregate_kernel, dim3(EE), dim3(256), 0, stream,
                       feat, ee, denom, src, dst, rst);

    if (resid)
        hipLaunchKernelGGL(finalize_res_kernel, dim3((NF + 255) / 256),
                           dim3(256), 0, stream, rst, resid, bias, NF);
    else
        hipLaunchKernelGGL(finalize_elu_kernel, dim3((NF + 255) / 256),
                           dim3(256), 0, stream, rst, bias, NF);
}

extern "C" void kernel_launch(void* const* d_in, const int* in_sizes, int n_in,
                              void* d_out, int out_size, void* d_ws, size_t ws_size,
                              hipStream_t stream)
{
    const float* x       = (const float*)d_in[0];
    const int*   pos_src = (const int*)d_in[1];
    const int*   pos_dst = (const int*)d_in[2];
    const int*   neg_src = (const int*)d_in[3];
    const int*   neg_dst = (const int*)d_in[4];
    const float* W0      = (const float*)d_in[5];
    const float* al0     = (const float*)d_in[6];
    const float* ar0     = (const float*)d_in[7];
    const float* b0      = (const float*)d_in[8];
    const float* W1      = (const float*)d_in[9];
    const float* al1     = (const float*)d_in[10];
    const float* ar1     = (const float*)d_in[11];
    const float* b1      = (const float*)d_in[12];
    float* out = (float*)d_out;

    // workspace layout (floats)
    float* ws = (float*)d_ws;
    size_t o = 0;
    float* feat = ws + o; o += (size_t)NN * FF;   // per-layer transformed feats
    float* hmid = ws + o; o += (size_t)NN * FF;   // layer-0 output (residual src)
    float* hfin = ws + o; o += (size_t)NN * FF;   // layer-1 output
    float* el   = ws + o; o += (size_t)NN * HH;
    float* er   = ws + o; o += (size_t)NN * HH;
    float* emax = ws + o; o += (size_t)NN * HH;
    float* den  = ws + o; o += (size_t)NN * HH;
    float* ee   = ws + o; o += (size_t)EE * HH;
    (void)ws_size; (void)in_sizes; (void)n_in; (void)out_size;

    // layer 0: x -> hmid   (ELU activation, no residual)
    run_gat_layer(stream, x, W0, al0, ar0, b0, pos_src, pos_dst,
                  feat, el, er, emax, den, ee, hmid, nullptr);

    // layer 1: hmid -> hfin  (identity residual, no activation)
    run_gat_layer(stream, hmid, W1, al1, ar1, b1, pos_src, pos_dst,
                  feat, el, er, emax, den, ee, hfin, hmid);

    // edge scores: pos then neg, each [E,H] flattened
    const int EH = EE * HH;
    hipLaunchKernelGGL(score_kernel, dim3((EH + 7) / 8), dim3(256), 0, stream,
                       hfin, pos_src, pos_dst, out, EH);
    hipLaunchKernelGGL(score_kernel, dim3((EH + 7) / 8), dim3(256), 0, stream,
                       hfin, neg_src, neg_dst, out + EH, EH);
}